// ESNDriver_25082609008885
// MI455X (gfx1250) — compile-verified
//
#include <hip/hip_runtime.h>
#include <cstdint>

// Problem constants (from reference)
#define RES_DIM 8192
#define CHUNKS  16
#define NSE     1342177   // int(0.02 * 8192 * 8192)
#define LEAK    0.6f
#define BIAS    1.6f

// Tuning
#define SPLITS  64        // workgroups per chunk -> 1024 WGs total
#define BLOCK   256       // 8 wave32s per WG
#define PFDIST  (8 * BLOCK)

#if __has_builtin(__builtin_amdgcn_global_load_async_to_lds_b128)
#define HAVE_ASYNC_B128 1
#endif
#if __has_builtin(__builtin_amdgcn_s_wait_asynccnt)
#define HAVE_WAIT_ASYNC 1
#endif

// Pointer types matching the async-to-LDS builtin signature:
//   void (v4i addrspace(1)*, v4i addrspace(3)*, imm int, imm int)
typedef int v4i __attribute__((ext_vector_type(4)));
typedef __attribute__((address_space(1))) v4i g_v4i;
typedef __attribute__((address_space(3))) v4i l_v4i;

// ---------------------------------------------------------------------------
// Kernel 0: initialize the global accumulator with proj + bias
// (folds one operand of the final elementwise update into the init pass)
// ---------------------------------------------------------------------------
__global__ void esn_init(const float* __restrict__ proj, float* __restrict__ wx) {
    int i = blockIdx.x * blockDim.x + threadIdx.x;   // exactly CHUNKS*RES_DIM threads
    wx[i] = proj[i] + BIAS;
}

// ---------------------------------------------------------------------------
// Kernel 1: COO scatter. One WG handles a slice of one chunk's nonzeros.
// state chunk (32KB) lives in LDS (async global->LDS copy, ASYNCcnt);
// partial wx (32KB) accumulated with LDS float atomics (ds_add_f32);
// flushed with global_atomic_add_f32 into the L2-resident wx.
// COO stream is read with TH=NT so the 258MB single-use stream does not
// evict the reused state/wx working set from L2.
// ---------------------------------------------------------------------------
__global__ void __launch_bounds__(BLOCK) esn_scatter(
    const float* __restrict__ vals,
    const int*   __restrict__ rows,
    const int*   __restrict__ cols,
    const float* __restrict__ state,
    float*       __restrict__ wx)
{
    __shared__ float s_state[RES_DIM];   // 32 KB
    __shared__ float s_wx[RES_DIM];      // 32 KB

    const int chunk = blockIdx.x / SPLITS;
    const int split = blockIdx.x % SPLITS;
    const float* __restrict__ st = state + (size_t)chunk * RES_DIM;

    // zero partial accumulator
    for (int i = threadIdx.x; i < RES_DIM; i += BLOCK)
        s_wx[i] = 0.0f;

    // bring the chunk's state vector into LDS
#ifdef HAVE_ASYNC_B128
    for (int i = threadIdx.x; i < RES_DIM / 4; i += BLOCK) {
        __builtin_amdgcn_global_load_async_to_lds_b128(
            (g_v4i*)(st + 4 * i),
            (l_v4i*)(&s_state[4 * i]),
            /*offset=*/0, /*cpol=*/0);
    }
#ifdef HAVE_WAIT_ASYNC
    __builtin_amdgcn_s_wait_asynccnt(0);
#endif
#else
    for (int i = threadIdx.x; i < RES_DIM / 4; i += BLOCK) {
        float4 t = ((const float4*)st)[i];
        ((float4*)s_state)[i] = t;
    }
#endif
    __syncthreads();

    // this WG's slice of the chunk's nonzeros
    const size_t base = (size_t)chunk * NSE;
    const float* __restrict__ vp = vals + base;
    const int*   __restrict__ rp = rows + base;
    const int*   __restrict__ cp = cols + base;

    const int per = (NSE + SPLITS - 1) / SPLITS;
    const int lo  = split * per;
    const int hi  = (lo + per < NSE) ? (lo + per) : NSE;

    for (int i = lo + (int)threadIdx.x; i < hi; i += BLOCK) {
        // speculative prefetch of the streamed COO arrays (global_prefetch_b8);
        // OOB speculative prefetches are silently dropped by hardware
        __builtin_prefetch(vp + i + PFDIST, 0, 0);
        __builtin_prefetch(rp + i + PFDIST, 0, 0);
        __builtin_prefetch(cp + i + PFDIST, 0, 0);

        // single-use stream: non-temporal loads (TH=NT)
        float v = __builtin_nontemporal_load(vp + i);
        int   r = __builtin_nontemporal_load(rp + i);
        int   c = __builtin_nontemporal_load(cp + i);
        unsafeAtomicAdd(&s_wx[r], v * s_state[c]);   // ds_add_f32 (LDS atomic)
    }
    __syncthreads();

    // flush partial accumulator into the global (L2-resident) wx
    float* __restrict__ wxc = wx + (size_t)chunk * RES_DIM;
    for (int i = threadIdx.x; i < RES_DIM; i += BLOCK)
        unsafeAtomicAdd(&wxc[i], s_wx[i]);           // global_atomic_add_f32
}

// ---------------------------------------------------------------------------
// Kernel 2: fused ESN update: out = leak*tanh(wx) + (1-leak)*state
// (wx already contains Wr@s + proj + bias)
// ---------------------------------------------------------------------------
__global__ void esn_update(const float* __restrict__ wx,
                           const float* __restrict__ state,
                           float*       __restrict__ out)
{
    int i = blockIdx.x * blockDim.x + threadIdx.x;   // exactly CHUNKS*RES_DIM threads
    float r = LEAK * tanhf(wx[i]) + (1.0f - LEAK) * state[i];
    __builtin_nontemporal_store(r, out + i);
}

// ---------------------------------------------------------------------------
extern "C" void kernel_launch(void* const* d_in, const int* in_sizes, int n_in,
                              void* d_out, int out_size, void* d_ws, size_t ws_size,
                              hipStream_t stream)
{
    (void)in_sizes; (void)n_in; (void)out_size; (void)ws_size;

    const float* proj  = (const float*)d_in[0];  // (CHUNKS, RES_DIM) f32
    const float* state = (const float*)d_in[1];  // (CHUNKS, RES_DIM) f32
    const float* vals  = (const float*)d_in[2];  // (CHUNKS, NSE) f32
    const int*   rows  = (const int*)  d_in[3];  // (CHUNKS, NSE) i32
    const int*   cols  = (const int*)  d_in[4];  // (CHUNKS, NSE) i32
    float*       out   = (float*)d_out;          // (CHUNKS, RES_DIM) f32
    float*       wx    = (float*)d_ws;           // CHUNKS*RES_DIM f32 scratch (512 KB)

    const int nElem = CHUNKS * RES_DIM;          // 131072

    esn_init   <<<nElem / 256,      256,   0, stream>>>(proj, wx);
    esn_scatter<<<CHUNKS * SPLITS,  BLOCK, 0, stream>>>(vals, rows, cols, state, wx);
    esn_update <<<nElem / 256,      256,   0, stream>>>(wx, state, out);
}